// DeltaQuantization_6236292513855
// MI455X (gfx1250) — compile-verified
//
#include <hip/hip_runtime.h>

#define NBINS 256
#define STEP 32
#define TROW 4096
#define NSTEPS (TROW / STEP)   // 128

typedef float v2f __attribute__((ext_vector_type(2)));
typedef float v4f __attribute__((ext_vector_type(4)));
typedef float v8f __attribute__((ext_vector_type(8)));

#if defined(__has_builtin) && __has_builtin(__builtin_amdgcn_exp2f)
#define EXP2F(x) __builtin_amdgcn_exp2f(x)
#else
#define EXP2F(x) exp2f(x)
#endif
#if defined(__has_builtin) && __has_builtin(__builtin_amdgcn_rcpf)
#define RCPF(x) __builtin_amdgcn_rcpf(x)
#else
#define RCPF(x) (1.0f / (x))
#endif

// ---- wave32 cross-lane helpers -------------------------------------------
// DPP16 ROW_XMASK (0x160|mask): pure-VALU xor-shuffle within each 16-lane
// row. old=0 + bound_ctrl=1 so no tie-copy mov is needed (all row lanes are
// always active here).
template <int XMASK>
__device__ __forceinline__ int dpp_xmask_i(int v) {
    return __builtin_amdgcn_update_dpp(0, v, 0x160 | XMASK, 0xf, 0xf, true);
}
template <int XMASK>
__device__ __forceinline__ float dpp_xmask_f(float v) {
    return __int_as_float(dpp_xmask_i<XMASK>(__float_as_int(v)));
}
__device__ __forceinline__ float row_sum16(float v) {
    v += dpp_xmask_f<1>(v);
    v += dpp_xmask_f<2>(v);
    v += dpp_xmask_f<4>(v);
    v += dpp_xmask_f<8>(v);
    return v;
}
__device__ __forceinline__ float row_sum8(float v) {   // sum within 8-lane banks
    v += dpp_xmask_f<1>(v);
    v += dpp_xmask_f<2>(v);
    v += dpp_xmask_f<4>(v);
    return v;
}
// Min over the 16-lane row for NON-NEGATIVE floats: IEEE bit patterns of
// non-negative floats are order-isomorphic to unsigned ints, so v_min_u32
// works and avoids the v_max canonicalize ops LLVM inserts for minnum.
__device__ __forceinline__ float row_min16_pos(float vf) {
    unsigned v = (unsigned)__float_as_int(vf);
    unsigned o;
    o = (unsigned)dpp_xmask_i<1>((int)v); v = v < o ? v : o;
    o = (unsigned)dpp_xmask_i<2>((int)v); v = v < o ? v : o;
    o = (unsigned)dpp_xmask_i<4>((int)v); v = v < o ? v : o;
    o = (unsigned)dpp_xmask_i<8>((int)v); v = v < o ? v : o;
    return __int_as_float((int)v);
}

// Split workgroup barrier that only drains DScnt (the barrier protocol is
// LDS-only). Avoids __syncthreads()'s s_wait_storecnt drain, which would put
// the streaming NT output stores onto the 128-step sequential critical path.
__device__ __forceinline__ void lds_barrier_signal() {
    asm volatile("s_wait_dscnt 0x0\n\ts_barrier_signal -1" ::: "memory");
}
__device__ __forceinline__ void lds_barrier_wait() {
    asm volatile("s_barrier_wait -1" ::: "memory");
}

// One block per batch row. 256 threads = 8 waves = 16 DPP rows ("halves").
// Half H = tid>>4 handles s in {2H, 2H+1}; lane l = tid&15 handles bins
// n in [16l, 16l+16). All softmax reductions are row-local.
__global__ __launch_bounds__(256, 1)
void dq_scan_kernel(const float* __restrict__ x,
                    const float* __restrict__ qbins,
                    const float* __restrict__ cscales,
                    const float* __restrict__ temp_p,
                    float* __restrict__ out) {
    __shared__ v4f xl4[TROW / 4];                 // staged x row (16 KB)
    __shared__ __align__(16) float pbuf[2][64];   // double-buffered partials

    const int b    = blockIdx.x;
    const int tid  = threadIdx.x;
    const int lane = tid & 31;
    const int H    = tid >> 4;           // 0..15
    const int l    = tid & 15;           // lane within DPP row

    const float temp = temp_p[0];
    const float kl2  = temp * 1.44269504088896340736f;   // temp * log2(e)

    // Stage full x row into LDS (coalesced, 4 x float4 per thread).
    const float* xrow = x + (size_t)b * TROW;
    __builtin_prefetch(xrow, 0, 3);                      // global_prefetch_b8
    const v4f* xr4 = (const v4f*)xrow;
#pragma unroll
    for (int i = 0; i < 4; ++i) xl4[tid + 256 * i] = xr4[tid + 256 * i];

    // Lane-private codebook slices: n = 16*l + j, j = 0..15.
    float q[16], cs[16];
    {
        const v4f* q4 = (const v4f*)qbins;
        const v4f* c4 = (const v4f*)cscales;
#pragma unroll
        for (int t = 0; t < 4; ++t) {
            v4f a = q4[l * 4 + t];
            v4f c = c4[l * 4 + t];
#pragma unroll
            for (int u = 0; u < 4; ++u) { q[4 * t + u] = a[u]; cs[4 * t + u] = c[u]; }
        }
    }
    __syncthreads();

    // A-operand index with the half-rotation trick: lanes 16-31 feed the
    // rotated partial pair into A's K={2,3} columns so one WMMA rowsum
    // yields the FULL 32-partial sum on every lane.
    const int idxA = l ^ ((lane & 16) >> 1);

    float pred = 0.0f;   // running predictor (uniform across block)
    float g    = 1.0f;   // cumulative codebook scale (uniform across block)

    float* outrow = out + (size_t)b * TROW * NBINS;
    const v2f* xl2 = (const v2f*)xl4;    // xi pairs: index stp*16 + H

    v2f xi = xl2[H];                     // software-pipelined xi for step 0

    for (int stp = 0; stp < NSTEPS; ++stp) {
        float s1 = 0.0f, s2 = 0.0f;
        v4f o[2][4];                     // staged quant for deferred stores

#pragma unroll
        for (int r = 0; r < 2; ++r) {
            const float delta = xi[r] - pred;

            float dist[16];
#pragma unroll
            for (int j = 0; j < 16; ++j) dist[j] = fabsf(delta - g * q[j]);

            // Balanced min3 tree (depth ~3), then row-wide u32-min DPP.
            const float m0 = fminf(fminf(dist[0],  dist[1]),  dist[2]);
            const float m1 = fminf(fminf(dist[3],  dist[4]),  dist[5]);
            const float m2 = fminf(fminf(dist[6],  dist[7]),  dist[8]);
            const float m3 = fminf(fminf(dist[9],  dist[10]), dist[11]);
            const float m4 = fminf(fminf(dist[12], dist[13]), dist[14]);
            const float ma = fminf(fminf(m0, m1), dist[15]);
            const float mb = fminf(fminf(m2, m3), m4);
            const float mind  = row_min16_pos(fminf(ma, mb));
            const float shift = kl2 * mind;

            float e[16];
#pragma unroll
            for (int j = 0; j < 16; ++j)
                e[j] = EXP2F(__builtin_fmaf(-kl2, dist[j], shift));

            // Balanced sum tree (depth 4), then row-wide DPP sum.
            const float t0 = (e[0] + e[1]) + (e[2] + e[3]);
            const float t1 = (e[4] + e[5]) + (e[6] + e[7]);
            const float t2 = (e[8] + e[9]) + (e[10] + e[11]);
            const float t3 = (e[12] + e[13]) + (e[14] + e[15]);
            const float denom = row_sum16((t0 + t1) + (t2 + t3));
            const float inv   = RCPF(denom);

            // Normalize; accumulate S1/S2 with 4 parallel FMA chains each.
            float a0 = 0.f, a1 = 0.f, a2 = 0.f, a3 = 0.f;
            float b0 = 0.f, b1 = 0.f, b2 = 0.f, b3 = 0.f;
#pragma unroll
            for (int j = 0; j < 16; j += 4) {
                const float q0 = e[j] * inv, q1 = e[j + 1] * inv;
                const float q2 = e[j + 2] * inv, q3 = e[j + 3] * inv;
                a0 = __builtin_fmaf(q0, q[j], a0);
                a1 = __builtin_fmaf(q1, q[j + 1], a1);
                a2 = __builtin_fmaf(q2, q[j + 2], a2);
                a3 = __builtin_fmaf(q3, q[j + 3], a3);
                b0 = __builtin_fmaf(q0, cs[j], b0);
                b1 = __builtin_fmaf(q1, cs[j + 1], b1);
                b2 = __builtin_fmaf(q2, cs[j + 2], b2);
                b3 = __builtin_fmaf(q3, cs[j + 3], b3);
                v4f ov; ov[0] = q0; ov[1] = q1; ov[2] = q2; ov[3] = q3;
                o[r][j >> 2] = ov;
            }
            s1 += (a0 + a1) + (a2 + a3);
            s2 += (b0 + b1) + (b2 + b3);
        }

        // Reduce only within 8-lane banks; the WMMA finishes the job.
        s1 = row_sum8(s1);
        s2 = row_sum8(s2);
        const int bufi = stp & 1;
        if ((l & 7) == 0) {
            const int i2 = 2 * H + (l >> 3);
            pbuf[bufi][i2]      = s1;    // 32 S1 partials at 0..31
            pbuf[bufi][32 + i2] = s2;    // 32 S2 partials at 32..63
        }

        // Signal early; hide barrier latency behind the streaming stores and
        // the next step's xi load, then wait.
        lds_barrier_signal();

        float* op = outrow + (size_t)(stp * STEP + 2 * H) * NBINS + l * 16;
#pragma unroll
        for (int r = 0; r < 2; ++r)
#pragma unroll
            for (int t = 0; t < 4; ++t)
                __builtin_nontemporal_store(o[r][t], (v4f*)(op + r * NBINS + 4 * t));

        const v2f xi_next = xl2[((stp + 1) & (NSTEPS - 1)) * (STEP / 2) + H];

        lds_barrier_wait();

        float S1, S2;
#if defined(__has_builtin) && __has_builtin(__builtin_amdgcn_wmma_f32_16x16x4_f32)
        {
            // A (16x4 f32, documented layout): lanes 0-15 VGPR{0,1} are
            // K={0,1}; lanes 16-31 VGPR{0,1} are K={2,3}. Load pair
            // {P2[2*idxA], P2[2*idxA+1]} so D[m,n] = P2[2m] + P2[2m+1] +
            // P2[2(m^8)] + P2[2(m^8)+1]; each lane's 8-register D sum is the
            // full 32-partial sum. B = all-ones (layout-invariant).
            const v2f* pb2 = (const v2f*)&pbuf[bufi][0];
            v2f A1 = pb2[idxA];
            v2f A2 = pb2[16 + idxA];
            v2f Bn; Bn[0] = 1.0f; Bn[1] = 1.0f;
            v8f Cz = {0.f, 0.f, 0.f, 0.f, 0.f, 0.f, 0.f, 0.f};
            v8f D1 = __builtin_amdgcn_wmma_f32_16x16x4_f32(
                false, A1, false, Bn, (short)0, Cz, false, false);
            v8f D2 = __builtin_amdgcn_wmma_f32_16x16x4_f32(
                false, A2, false, Bn, (short)0, Cz, false, false);
            // Pairwise order maps to v_pk_add_f32 on consecutive VGPR pairs.
            S1 = ((D1[0] + D1[2]) + (D1[4] + D1[6])) +
                 ((D1[1] + D1[3]) + (D1[5] + D1[7]));
            S2 = ((D2[0] + D2[2]) + (D2[4] + D2[6])) +
                 ((D2[1] + D2[3]) + (D2[5] + D2[7]));
        }
#else
        {
            S1 = 0.0f; S2 = 0.0f;
#pragma unroll
            for (int i = 0; i < 32; ++i) { S1 += pbuf[bufi][i]; S2 += pbuf[bufi][32 + i]; }
        }
#endif

        pred = __builtin_fmaf(g * S1, 1.0f / 32.0f, pred);   // + mean soft-quantized delta
        g    = g * (S2 * (1.0f / 32.0f));                    // adaptive codebook rescale
        xi   = xi_next;
    }
}

extern "C" void kernel_launch(void* const* d_in, const int* in_sizes, int n_in,
                              void* d_out, int out_size, void* d_ws, size_t ws_size,
                              hipStream_t stream) {
    const float* x     = (const float*)d_in[0];
    const float* qbins = (const float*)d_in[1];
    const float* cscal = (const float*)d_in[2];
    const float* temp  = (const float*)d_in[3];
    float* out = (float*)d_out;

    const int B = in_sizes[0] / TROW;   // 32
    dq_scan_kernel<<<dim3(B), dim3(256), 0, stream>>>(x, qbins, cscal, temp, out);
}